// Tokenizer_68719476856
// MI455X (gfx1250) — compile-verified
//
#include <hip/hip_runtime.h>
#include <hip/hip_bf16.h>
#include <stdint.h>

// Problem geometry (from the reference)
#define BATCH 32
#define NWIN  1020
#define FLATN 128
#define VOCAB 2048
#define EDIM  512
#define NWINDOWS (BATCH * NWIN)          // 32640 windows
#define WAVES_PER_BLOCK 8
#define BLOCK_THREADS (WAVES_PER_BLOCK * 32)

// pw[f] = P^(128-f) mod 2^32  (only low 32 bits of the wrapping int64 hash
// are needed because the final mod 2048 divides 2^32)
struct PowTable { uint32_t v[FLATN]; };

__host__ __device__ constexpr PowTable make_pow_table() {
    PowTable t{};
    const uint32_t P = (uint32_t)(17461204521323ull & 0xffffffffull);
    uint32_t p = P;                       // P^1
    for (int e = 1; e <= FLATN; ++e) {    // v[128-e] = P^e
        t.v[FLATN - e] = p;
        p *= P;                           // wraps mod 2^32 — exactly what we want
    }
    return t;
}

__device__ __constant__ PowTable g_pow = make_pow_table();

__global__ __launch_bounds__(BLOCK_THREADS)
void tokenizer_hash_gather_kernel(const long long* __restrict__ fuzz,
                                  const float* __restrict__ emb,
                                  float* __restrict__ out)
{
    // 2 KB staging row per wave; async DMA target must be 16B aligned
    __shared__ __align__(16) float smem[WAVES_PER_BLOCK * EDIM];

    const int lane = threadIdx.x & 31;
    const int wid  = threadIdx.x >> 5;
    const int gw   = blockIdx.x * WAVES_PER_BLOCK + wid;   // window id
    if (gw >= NWINDOWS) return;                            // exact divide; uniform

    // ---------------- polynomial hash (closed form, 32-bit wrap) -------------
    // Lane handles tokens f = 4*lane .. 4*lane+3 of this window.
    const long long* fz = fuzz + (size_t)gw * FLATN + 4 * lane;
    longlong2 t01 = *(const longlong2*)(fz + 0);
    longlong2 t23 = *(const longlong2*)(fz + 2);

    const int f0 = 4 * lane;
    uint32_t s = 0;
    s += (uint32_t)(t01.x + 1) * g_pow.v[f0 + 0];
    s += (uint32_t)(t01.y + 1) * g_pow.v[f0 + 1];
    s += (uint32_t)(t23.x + 1) * g_pow.v[f0 + 2];
    s += (uint32_t)(t23.y + 1) * g_pow.v[f0 + 3];

    // wave32 butterfly reduction -> every lane holds the full sum
    #pragma unroll
    for (int m = 16; m > 0; m >>= 1) s += __shfl_xor(s, m, 32);

    const uint32_t tok = s & (uint32_t)(VOCAB - 1);        // floored mod 2048

    // ---------------- gather row via async global<->LDS DMA ------------------
    // LDS byte address = low 32 bits of the generic pointer to __shared__
    const uint32_t ldsRow  = (uint32_t)(uintptr_t)(&smem[wid * EDIM]);
    const uint32_t ldsOff  = ldsRow + (uint32_t)lane * 16u;
    const uint32_t srcOff  = tok * (uint32_t)(EDIM * 4) + (uint32_t)lane * 16u;
    const uint32_t dstOff  = (uint32_t)gw * (uint32_t)(EDIM * 4) + (uint32_t)lane * 16u;
    const uint64_t embBase = (uint64_t)(uintptr_t)emb;
    const uint64_t outBase = (uint64_t)(uintptr_t)out;

    // 32 lanes x 16B = 512B per issue; 4 issues cover the 2048B row.
    #pragma unroll
    for (int j = 0; j < 4; ++j) {
        asm volatile("global_load_async_to_lds_b128 %0, %1, %2"
                     :
                     : "v"(ldsOff + (uint32_t)(j * 512)),
                       "v"(srcOff + (uint32_t)(j * 512)),
                       "s"(embBase)
                     : "memory");
    }
    asm volatile("s_wait_asynccnt 0" ::: "memory");

    #pragma unroll
    for (int j = 0; j < 4; ++j) {
        asm volatile("global_store_async_from_lds_b128 %0, %1, %2"
                     :
                     : "v"(dstOff + (uint32_t)(j * 512)),
                       "v"(ldsOff + (uint32_t)(j * 512)),
                       "s"(outBase)
                     : "memory");
    }
    asm volatile("s_wait_asynccnt 0" ::: "memory");
}

extern "C" void kernel_launch(void* const* d_in, const int* in_sizes, int n_in,
                              void* d_out, int out_size, void* d_ws, size_t ws_size,
                              hipStream_t stream) {
    (void)in_sizes; (void)n_in; (void)out_size; (void)d_ws; (void)ws_size;
    const long long* fuzz = (const long long*)d_in[0];   // int64 [32,1020,128]
    const float*     emb  = (const float*)d_in[1];       // f32   [2048,512]
    float*           out  = (float*)d_out;               // f32   [32,1020,512]

    dim3 grid((NWINDOWS + WAVES_PER_BLOCK - 1) / WAVES_PER_BLOCK);  // 4080
    dim3 block(BLOCK_THREADS);                                       // 256 = 8 waves
    tokenizer_hash_gather_kernel<<<grid, block, 0, stream>>>(fuzz, emb, out);
}